// BSpline_80264348827981
// MI455X (gfx1250) — compile-verified
//
#include <hip/hip_runtime.h>

// B-spline basis (Cox-de Boor, order 7, 16 knots, 8 basis fns) + row contraction.
// out[n,k] = sum_h x[n,h] * N_k(x[n,h]),  n in [0,131072), k,h in [0,8).
//
// Strategy:
//  - setup kernel precomputes per-(p,i) reciprocal tables A=1/(U[i+p]-U[i]),
//    B=U[i]*A, C=U[i+p]*A into d_ws (352 floats). Zero-denominator -> 0,
//    matching the reference's jnp.where semantics exactly.
//  - main kernel: one thread per row; x staged global->LDS via CDNA5 async
//    copy (ASYNCcnt path); recursion is pure FMA with wave-uniform scalar
//    operands; two b128 stores per row.

#define ORDER_P 7
#define GRID_K 8
#define NKNOT 16           // GRID + ORDER + 1
#define NSEG (NKNOT - 1)   // 15 degree-0 intervals
#define HDIM 8
#define ROWS_PER_BLOCK 256

#if __has_builtin(__builtin_amdgcn_global_load_async_to_lds_b128) && \
    __has_builtin(__builtin_amdgcn_s_wait_asynccnt)
#define USE_ASYNC_LDS 1
#else
#define USE_ASYNC_LDS 0
#endif

#if USE_ASYNC_LDS
typedef int v4i_t __attribute__((__vector_size__(16)));
typedef __attribute__((address_space(1))) v4i_t* async_gptr_t;
typedef __attribute__((address_space(3))) v4i_t* async_lptr_t;
#endif

__global__ void bspline_setup(const float* __restrict__ knots,
                              float* __restrict__ tab) {
  const int t = threadIdx.x;
  if (t < NKNOT) tab[t] = knots[t];
  const int p = t / 16 + 1;
  const int i = t % 16;
  if (p <= ORDER_P && (i + p) < NKNOT) {
    const float d = knots[i + p] - knots[i];
    const float inv = (d != 0.0f) ? (1.0f / d) : 0.0f;
    tab[16 + (p - 1) * 16 + i] = inv;                      // A
    tab[16 + 112 + (p - 1) * 16 + i] = knots[i] * inv;     // B = U[i]*A
    tab[16 + 224 + (p - 1) * 16 + i] = knots[i + p] * inv; // C = U[i+p]*A
  }
}

__global__ __launch_bounds__(ROWS_PER_BLOCK) void bspline_fused(
    const float* __restrict__ x,
    const float* __restrict__ tab,
    float* __restrict__ out) {
  __shared__ __align__(16) float tile[ROWS_PER_BLOCK * HDIM];
  const int t = threadIdx.x;
  const long long base = (long long)blockIdx.x * (ROWS_PER_BLOCK * HDIM);
  const float* gx = x + base;

#if USE_ASYNC_LDS
  {
    // Two b128 async copies per lane: 256 lanes * 32B = 8KB tile.
    unsigned long long g0 = (unsigned long long)(gx + t * 4);
    unsigned long long g1 = (unsigned long long)(gx + 1024 + t * 4);
    unsigned l0 = (unsigned)(unsigned long long)(&tile[t * 4]);
    unsigned l1 = (unsigned)(unsigned long long)(&tile[1024 + t * 4]);
    __builtin_amdgcn_global_load_async_to_lds_b128(
        (async_gptr_t)g0, (async_lptr_t)l0, 0, 0);
    __builtin_amdgcn_global_load_async_to_lds_b128(
        (async_gptr_t)g1, (async_lptr_t)l1, 0, 0);
    __builtin_amdgcn_s_wait_asynccnt(0);
  }
#else
  {
    const float4* g4 = (const float4*)gx;
    float4* s4 = (float4*)tile;
    s4[t] = g4[t];
    s4[t + ROWS_PER_BLOCK] = g4[t + ROWS_PER_BLOCK];
  }
#endif
  __syncthreads();

  // Wave-uniform tables (compiler should select s_load for these).
  float cU[NKNOT];
#pragma unroll
  for (int i = 0; i < NKNOT; ++i) cU[i] = tab[i];
  const float* __restrict__ A = tab + 16;
  const float* __restrict__ Bm = tab + 16 + 112;
  const float* __restrict__ Cp = tab + 16 + 224;

  float acc[GRID_K];
#pragma unroll
  for (int k = 0; k < GRID_K; ++k) acc[k] = 0.0f;

#pragma unroll 1
  for (int h = 0; h < HDIM; ++h) {
    const float u = tile[t * HDIM + h];
    float N[NSEG];
    // Degree 0: half-open interval indicator, matching the reference.
#pragma unroll
    for (int i = 0; i < NSEG; ++i)
      N[i] = (cU[i] <= u && u < cU[i + 1]) ? 1.0f : 0.0f;
    // Levels p = 1..7:  N[i] <- w1*N[i] + w2*N[i+1]
    //   w1 = (u-U[i])*A[p][i]       = fma(u, A,  -B)
    //   w2 = (U[i+p+1]-u)*A[p][i+1] = fma(-u, A', C')
#pragma unroll
    for (int p = 1; p <= ORDER_P; ++p) {
#pragma unroll
      for (int i = 0; i < NSEG - p; ++i) {
        const int r = (p - 1) * 16;
        const float w1 = fmaf(u, A[r + i], -Bm[r + i]);
        const float w2 = fmaf(-u, A[r + i + 1], Cp[r + i + 1]);
        N[i] = fmaf(w2, N[i + 1], w1 * N[i]);
      }
    }
#pragma unroll
    for (int k = 0; k < GRID_K; ++k) acc[k] = fmaf(u, N[k], acc[k]);
  }

  float4* o4 = (float4*)(out + base);
  o4[t * 2] = make_float4(acc[0], acc[1], acc[2], acc[3]);
  o4[t * 2 + 1] = make_float4(acc[4], acc[5], acc[6], acc[7]);
}

extern "C" void kernel_launch(void* const* d_in, const int* in_sizes, int n_in,
                              void* d_out, int out_size, void* d_ws,
                              size_t ws_size, hipStream_t stream) {
  (void)n_in; (void)out_size; (void)ws_size;
  const float* x = (const float*)d_in[0];
  const float* knots = (const float*)d_in[1];
  float* out = (float*)d_out;
  float* tab = (float*)d_ws;  // 352 floats used

  hipLaunchKernelGGL(bspline_setup, dim3(1), dim3(128), 0, stream, knots, tab);

  const int rows = in_sizes[0] / HDIM;      // 131072
  const int nblk = rows / ROWS_PER_BLOCK;   // 512
  hipLaunchKernelGGL(bspline_fused, dim3(nblk), dim3(ROWS_PER_BLOCK), 0, stream,
                     x, tab, out);
}